// Difference3DCostVolume_51462298140770
// MI455X (gfx1250) — compile-verified
//
#include <hip/hip_runtime.h>
#include <stdint.h>

// Problem constants (reference: N, C, H, W = 2, 32, 128, 256; MAX_DISP = 48)
#define DIM_N 2
#define DIM_C 32
#define DIM_H 128
#define DIM_W 256
#define DIM_D 48

typedef unsigned int u32x4 __attribute__((ext_vector_type(4)));
typedef int          i32x4 __attribute__((ext_vector_type(4)));
typedef int          i32x8 __attribute__((ext_vector_type(8)));
typedef float        f32x4 __attribute__((ext_vector_type(4)));

// Issue one TDM descriptor: copy `nelem` contiguous f32 from global -> LDS.
// D# layout per CDNA5 ISA ch.8:
//   group0: [1:0]count=1 | [63:32]lds_addr | [120:64]global_addr | [127:126]type=2
//   group1: [17:16]data_size=2(4B) | [79:48]tensor_dim0 | [111:80]tensor_dim1=1 |
//           [127:112]tile_dim0 | tile_dim1/2=0 (1-D tile) | [207:160]dim0_stride
__device__ __forceinline__ void tdm_load_row_f32(const float* gsrc,
                                                 uint32_t lds_byte_addr,
                                                 int nelem) {
  const uint64_t ga = (uint64_t)(uintptr_t)gsrc;
  u32x4 g0;
  g0[0] = 1u;                                            // count=1 (valid user desc)
  g0[1] = lds_byte_addr;                                 // lds_addr
  g0[2] = (uint32_t)(ga & 0xFFFFFFFFu);                  // global_addr[31:0]
  g0[3] = (uint32_t)((ga >> 32) & 0x1FFFFFFu)            // global_addr[56:32]
        | (2u << 30);                                    // type = 2 ("image")
  i32x8 g1;
  g1[0] = (int)(2u << 16);                               // data_size=2 -> 4 bytes; wg_mask=0
  g1[1] = (int)(((uint32_t)nelem & 0xFFFFu) << 16);      // tensor_dim0[15:0] @ bits 63:48
  g1[2] = (int)((((uint32_t)nelem >> 16) & 0xFFFFu)      // tensor_dim0[31:16]
        | (1u << 16));                                   // tensor_dim1 = 1
  g1[3] = (int)(((uint32_t)nelem & 0xFFFFu) << 16);      // tile_dim0 @ bits 127:112
  g1[4] = 0;                                             // tile_dim1=0, tile_dim2=0 (1-D)
  g1[5] = nelem;                                         // tensor_dim0_stride[31:0]
  g1[6] = 0;                                             // stride hi / dim1_stride lo
  g1[7] = 0;                                             // dim1_stride hi
  const i32x4 gz4 = {0, 0, 0, 0};
#if __clang_major__ >= 23
  const i32x8 gz8 = {0, 0, 0, 0, 0, 0, 0, 0};
  __builtin_amdgcn_tensor_load_to_lds(g0, g1, gz4, gz4, gz8, 0);
#else
  __builtin_amdgcn_tensor_load_to_lds(g0, g1, gz4, gz4, 0);
#endif
}

// One block per (n,c,h) row. TDM stages the 1 KB l-row and r-row into LDS,
// then 256 threads emit all 48 disparities as NT b128 stores.
// Lane mapping: w4 = (tid & 63)*4 (float4 column), dlane = tid >> 6 (wave-uniform),
// so each 32-lane wave writes 512 contiguous bytes per store.
__global__ void __launch_bounds__(256)
cost_volume_kernel(const float* __restrict__ L, const float* __restrict__ R,
                   float* __restrict__ out) {
  __shared__ __align__(16) float sL[DIM_W];
  __shared__ __align__(16) float sR[DIM_W];

  const int row = blockIdx.x;          // (n*C + c)*H + h
  const int h   = row & (DIM_H - 1);
  const int nc  = row >> 7;            // row / H, H == 128
  const float* lrow = L + (size_t)row * DIM_W;
  const float* rrow = R + (size_t)row * DIM_W;

  if (threadIdx.x == 0) {
    tdm_load_row_f32(lrow, (uint32_t)(uintptr_t)&sL[0], DIM_W);
    tdm_load_row_f32(rrow, (uint32_t)(uintptr_t)&sR[0], DIM_W);
    __builtin_amdgcn_s_wait_tensorcnt(0);  // data committed to LDS
  }
  __syncthreads();

  const int t     = threadIdx.x;
  const int w0    = (t & 63) << 2;     // float4 start column
  const int dlane = t >> 6;            // 0..3, uniform within a wave
  const f32x4 l4  = *(const f32x4*)&sL[w0];

  float* obase = out + (((size_t)nc * DIM_D) * DIM_H + h) * DIM_W + w0;

#pragma unroll
  for (int i = 0; i < DIM_D / 4; ++i) {
    const int d = i * 4 + dlane;
    f32x4 o;
#pragma unroll
    for (int j = 0; j < 4; ++j) {
      const int ww = w0 + j;
      const int ri = ww - d;
      const float rv = sR[ri < 0 ? 0 : ri];     // clamped; unused when masked
      o[j] = (ww >= d) ? (l4[j] - rv) : 1.0f;   // init value 1.0 where w < d
    }
    // 403 MB stream that can never live in L2 -> non-temporal b128 stores.
    __builtin_nontemporal_store(o, (f32x4*)(obase + (size_t)d * (DIM_H * DIM_W)));
  }
}

extern "C" void kernel_launch(void* const* d_in, const int* in_sizes, int n_in,
                              void* d_out, int out_size, void* d_ws, size_t ws_size,
                              hipStream_t stream) {
  (void)in_sizes; (void)n_in; (void)out_size; (void)d_ws; (void)ws_size;
  const float* L = (const float*)d_in[0];
  const float* R = (const float*)d_in[1];
  float* out = (float*)d_out;

  const int nblocks = DIM_N * DIM_C * DIM_H;  // 8192 rows
  cost_volume_kernel<<<nblocks, 256, 0, stream>>>(L, R, out);
}